// ARONetModel_25168508354975
// MI455X (gfx1250) — compile-verified
//
#include <hip/hip_runtime.h>

// ARONet cone-cast top-k + relative features for MI455X (gfx1250, wave32).
//
//  - One block per (batch, anchor, 128-query tile). Phase 1 precomputes
//    anchor->point directions + distances for all 2048 points into LDS
//    (amortized over 128 queries).
//  - Phase 2: each wave owns 16 queries. The cos(dir_aq, dir_ap) einsum has
//    K=3 -> V_WMMA_F32_16X16X4_F32 (A = 16 points x 4, B = 4 x 16 queries,
//    zero-padded K). 128 WMMA tiles cover the 2048 points.
//  - D layout (VGPR v, lane l -> M = v + 8*(l>=16), N = l&15): each lane owns
//    one query and 8/16 points per tile; register-resident sorted top-16
//    (dist, idx), maintained with a fully branch-free select network behind a
//    rare-path guard (cone hit rate ~1.7%).
//  - Lane pairs (l, l+16) merge their halves via __shfl + bitonic
//    half-cleaner + 4-stage bitonic merge network (all v_cndmask selects).
//  - +inf slots emit (0,0,0,0), matching the reference's hit=qry padding.

typedef float v2f __attribute__((ext_vector_type(2)));
typedef float v8f __attribute__((ext_vector_type(8)));

#define NB 2
#define NP 2048
#define NQ 512
#define NA 48
#define NL 16
#define QPB 128                 // queries per block = 8 waves * 16
#define QTILES (NQ / QPB)       // 4

__device__ __forceinline__ float finf() { return __builtin_inff(); }

__global__ __launch_bounds__(256) void aronet_cone_topk_kernel(
    const float* __restrict__ pcd, const float* __restrict__ qry,
    const float* __restrict__ anc, float* __restrict__ out)
{
    constexpr float COS_TH = 0.96592582628906828675f;  // cos(pi/12)

    __shared__ float4 s_dir[NP];   // (dir.x, dir.y, dir.z, dist) per point
    __shared__ float  s_dist[NP];  // dist only, for vectorized b128 re-reads

    const int bid = blockIdx.x;
    const int qt  = bid % QTILES;
    const int a   = (bid / QTILES) % NA;
    const int b   = bid / (QTILES * NA);

    const float ax = anc[(b * NA + a) * 3 + 0];
    const float ay = anc[(b * NA + a) * 3 + 1];
    const float az = anc[(b * NA + a) * 3 + 2];

    // ---- Phase 1: anchor->point dir/dist tables (reused by 128 queries) ----
    for (int p = threadIdx.x; p < NP; p += blockDim.x) {
        const float px = pcd[(b * NP + p) * 3 + 0];
        const float py = pcd[(b * NP + p) * 3 + 1];
        const float pz = pcd[(b * NP + p) * 3 + 2];
        const float vx = px - ax, vy = py - ay, vz = pz - az;
        const float d   = sqrtf(vx * vx + vy * vy + vz * vz);
        const float inv = 1.0f / d;
        s_dir[p]  = make_float4(vx * inv, vy * inv, vz * inv, d);
        s_dist[p] = d;
    }
    __syncthreads();

    // ---- Phase 2: per-wave 16 queries, WMMA cos tiles + register top-k ----
    const int  lane = threadIdx.x & 31;
    const int  wave = threadIdx.x >> 5;
    const bool hi   = lane >= 16;
    const int  qn   = qt * QPB + wave * 16 + (lane & 15);

    const float qx = qry[(b * NQ + qn) * 3 + 0];
    const float qy = qry[(b * NQ + qn) * 3 + 1];
    const float qz = qry[(b * NQ + qn) * 3 + 2];
    const float vqx = qx - ax, vqy = qy - ay, vqz = qz - az;
    const float qmod = sqrtf(vqx * vqx + vqy * vqy + vqz * vqz);
    const float qinv = 1.0f / qmod;

    // B fragment (4x16, K x N=query): low lanes K=0,1; high lanes K=2,3(=0)
    v2f bfrag;
    bfrag.x = (hi ? vqz : vqx) * qinv;
    bfrag.y = hi ? 0.0f : vqy * qinv;

    float vals[NL];
    int   idxs[NL];
#pragma unroll
    for (int j = 0; j < NL; ++j) { vals[j] = finf(); idxs[j] = 0; }

    // lane-dependent component offset: low lanes read (x,y), high lanes (z,w)
    const float* dirbase = (const float*)&s_dir[0] + (hi ? 2 : 0);

    for (int t = 0; t < NP / 16; ++t) {
        const int pb = t * 16;
        const int mb = pb + (hi ? 8 : 0);  // this lane's 8 point rows

        // A fragment (16x4, M=point x K): one ds_load_b64 + one zero-select
        const float2 pdl = *(const float2*)(dirbase + 4 * (pb + (lane & 15)));
        v2f afrag;
        afrag.x = pdl.x;
        afrag.y = hi ? 0.0f : pdl.y;

        // this lane's 8 candidate distances: two ds_load_b128 (broadcast)
        const float4 dl = *(const float4*)&s_dist[mb];
        const float4 dh = *(const float4*)&s_dist[mb + 4];
        const float dists[8] = {dl.x, dl.y, dl.z, dl.w,
                                dh.x, dh.y, dh.z, dh.w};

        v8f c = {};
        v8f d = __builtin_amdgcn_wmma_f32_16x16x4_f32(
            /*neg_a=*/false, afrag, /*neg_b=*/false, bfrag,
            /*c_mod=*/(short)0, c, /*reuse_a=*/false, /*reuse_b=*/false);

#pragma unroll
        for (int v = 0; v < 8; ++v) {
            const float cand = (d[v] > COS_TH) ? dists[v] : finf();
            if (cand < vals[NL - 1]) {   // rare path: cone hit beating current max
                float cv = cand; int ci = mb + v;
                bool  ins = false;
#pragma unroll
                for (int j = 0; j < NL; ++j) {   // branch-free insertion
                    const float tv = vals[j]; const int ti = idxs[j];
                    const bool  sw = ins | (cv < tv);
                    vals[j] = sw ? cv : tv;
                    idxs[j] = sw ? ci : ti;
                    cv = sw ? tv : cv;
                    ci = sw ? ti : ci;
                    ins = sw;
                }
            }
        }
    }

    // ---- Merge lane-pair halves: bitonic half-cleaner + merge network ----
    float ov[NL]; int oi[NL];
#pragma unroll
    for (int i = 0; i < NL; ++i) {        // snapshot partner (reversed order)
        ov[i] = __shfl(vals[NL - 1 - i], lane ^ 16, 32);
        oi[i] = __shfl(idxs[NL - 1 - i], lane ^ 16, 32);
    }
#pragma unroll
    for (int i = 0; i < NL; ++i) {        // lower half of bitonic split
        const bool take = ov[i] < vals[i];
        vals[i] = take ? ov[i] : vals[i];
        idxs[i] = take ? oi[i] : idxs[i];
    }
#pragma unroll
    for (int s = 8; s > 0; s >>= 1) {     // sort the bitonic 16-sequence
#pragma unroll
        for (int i = 0; i < NL; ++i) {
            if ((i & s) == 0) {
                const int j = i + s;
                const float vi = vals[i], vj = vals[j];
                const int   ii = idxs[i], ij = idxs[j];
                const bool  sw = vj < vi;
                vals[i] = sw ? vj : vi;  vals[j] = sw ? vi : vj;
                idxs[i] = sw ? ij : ii;  idxs[j] = sw ? ii : ij;
            }
        }
    }

    // ---- Emit features: [b, q, a, 1+16, 4] ----
    if (lane < 16) {
        float4* out4 = (float4*)out;
        const size_t base =
            (((size_t)b * NQ + (size_t)qn) * NA + (size_t)a) * (NL + 1);

        out4[base] = make_float4(vqx, vqy, vqz, qmod);  // feat_anc2qry

#pragma unroll
        for (int k = 0; k < NL; ++k) {
            const bool pad = !(vals[k] < finf());   // no valid hit -> zeros
            const int  p   = idxs[k];               // 0 when padded (safe)
            const float hx = pcd[(b * NP + p) * 3 + 0];
            const float hy = pcd[(b * NP + p) * 3 + 1];
            const float hz = pcd[(b * NP + p) * 3 + 2];
            const float dx = hx - qx, dy = hy - qy, dz = hz - qz;
            const float sq  = dx * dx + dy * dy + dz * dz;
            const float mod = sqrtf(sq);
            out4[base + 1 + k] = make_float4(pad ? 0.0f : dx,
                                             pad ? 0.0f : dy,
                                             pad ? 0.0f : dz,
                                             pad ? 0.0f : mod);
        }
    }
}

extern "C" void kernel_launch(void* const* d_in, const int* in_sizes, int n_in,
                              void* d_out, int out_size, void* d_ws, size_t ws_size,
                              hipStream_t stream) {
    (void)in_sizes; (void)n_in; (void)out_size; (void)d_ws; (void)ws_size;
    const float* pcd = (const float*)d_in[0];
    const float* qry = (const float*)d_in[1];
    const float* anc = (const float*)d_in[2];
    float* out = (float*)d_out;

    dim3 grid(NB * NA * QTILES);   // 2 * 48 * 4 = 384 blocks
    dim3 block(256);               // 8 wave32 waves = 128 queries per block
    hipLaunchKernelGGL(aronet_cone_topk_kernel, grid, block, 0, stream,
                       pcd, qry, anc, out);
}